// SE3TransformerWrapper_65326452572272
// MI455X (gfx1250) — compile-verified
//
#include <hip/hip_runtime.h>
#include <math.h>

#define NN 20000
#define NE 640000
#define CCH 32

typedef float v2f __attribute__((ext_vector_type(2)));
typedef float v8f __attribute__((ext_vector_type(8)));

static __device__ __forceinline__ v8f zero8() {
  v8f z = {0.f,0.f,0.f,0.f,0.f,0.f,0.f,0.f};
  return z;
}

static __device__ __forceinline__ v8f wmma_f32(v2f a, v2f b, v8f c) {
  // (neg_a, A, neg_b, B, c_mod, C, reuse_a, reuse_b)
  return __builtin_amdgcn_wmma_f32_16x16x4_f32(false, a, false, b, (short)0, c, false, false);
}

static __device__ __forceinline__ void atomicMaxFloat(float* addr, float val) {
  // ordered-bit trick; addr must be initialized to -inf (0xFF800000)
  if (val >= 0.0f) atomicMax((int*)addr, __float_as_int(val));
  else             atomicMin((unsigned int*)addr, __float_as_uint(val));
}

// ---------------------------------------------------------------------------
// Zero / init per-node accumulators, -inf max, zero Z.
// ---------------------------------------------------------------------------
__global__ void prep_accum_kernel(float* __restrict__ mx, float* __restrict__ Z,
                                  float* __restrict__ a0,
                                  float* __restrict__ a1x, float* __restrict__ a1y,
                                  float* __restrict__ a1z, int n) {
  int i = blockIdx.x * blockDim.x + threadIdx.x;
  if (i >= n) return;
  mx[i] = __int_as_float(0xFF800000);
  Z[i]  = 0.f;
  size_t b = (size_t)i * CCH;
#pragma unroll
  for (int c = 0; c < CCH; ++c) {
    a0[b+c] = 0.f; a1x[b+c] = 0.f; a1y[b+c] = 0.f; a1z[b+c] = 0.f;
  }
}

// ---------------------------------------------------------------------------
// WMMA GEMM:  Y[M,NO] (+)= X[M,32] @ W[32,NO]
// K fixed = 32, NO<=32, M multiple of 16. 4 waves/block, 16 rows/wave.
// Weights stored TRANSPOSED in LDS (sWt[n][k]) so each B fragment
// (W[k0][n], W[k0+1][n]) is one aligned 8-byte LDS load.
// ---------------------------------------------------------------------------
__global__ __launch_bounds__(128)
void gemm_wmma_kernel(const float* __restrict__ X, const float* __restrict__ W,
                      float* __restrict__ Y, int M, int NO, int accum) {
  __shared__ float sWt[32][34]; // [n][k]
  int tid = threadIdx.x;
  for (int idx = tid; idx < 32 * 32; idx += blockDim.x) {
    int n = idx >> 5, k = idx & 31;
    sWt[n][k] = (n < NO) ? W[k * NO + n] : 0.f;
  }
  __syncthreads();

  int wave = tid >> 5, lane = tid & 31;
  int r0 = (blockIdx.x * (blockDim.x >> 5) + wave) << 4;
  if (r0 >= M) return;

  int m = lane & 15, kh = lane >> 4;
  v8f acc0 = zero8();
  v8f acc1 = zero8();

#pragma unroll
  for (int kb = 0; kb < 32; kb += 4) {
    int k0 = kb + 2 * kh;
    v2f a = *(const v2f*)(X + (size_t)(r0 + m) * 32 + k0);
    v2f b0 = *(const v2f*)&sWt[m][k0];
    v2f b1 = *(const v2f*)&sWt[16 + m][k0];
    acc0 = wmma_f32(a, b0, acc0);
    acc1 = wmma_f32(a, b1, acc1);
  }

  {
    int n = m;
    if (n < NO) {
#pragma unroll
      for (int r = 0; r < 8; ++r) {
        int row = r0 + r + (kh << 3);
        size_t o = (size_t)row * NO + n;
        float v = acc0[r];
        if (accum) v += Y[o];
        Y[o] = v;
      }
    }
  }
  {
    int n = 16 + m;
    if (n < NO) {
#pragma unroll
      for (int r = 0; r < 8; ++r) {
        int row = r0 + r + (kh << 3);
        size_t o = (size_t)row * NO + n;
        float v = acc1[r];
        if (accum) v += Y[o];
        Y[o] = v;
      }
    }
  }
}

// ---------------------------------------------------------------------------
// Edge pass 1: attention logits + segment max (atomic).
// ---------------------------------------------------------------------------
__global__ void edge_logits_kernel(const int* __restrict__ src, const int* __restrict__ dst,
                                   const float* __restrict__ qn, const float* __restrict__ kn,
                                   float* __restrict__ logit, float* __restrict__ mx, int E) {
  int i = blockIdx.x * blockDim.x + threadIdx.x;
  if (i >= E) return;
  int s = src[i], t = dst[i];
  const float4* q = (const float4*)(qn + (size_t)t * CCH);
  const float4* k = (const float4*)(kn + (size_t)s * CCH);
  float acc = 0.f;
#pragma unroll
  for (int j = 0; j < 8; ++j) {
    float4 a = q[j], b = k[j];
    acc += a.x * b.x + a.y * b.y + a.z * b.z + a.w * b.w;
  }
  float lg = acc * 0.17677669529663687f; // 1/sqrt(32)
  logit[i] = lg;
  atomicMaxFloat(mx + t, lg);
}

// ---------------------------------------------------------------------------
// Edge pass 2: exp(logit - max) and segment sum (atomic).
// ---------------------------------------------------------------------------
__global__ void edge_expsum_kernel(const int* __restrict__ dst,
                                   const float* __restrict__ logit, const float* __restrict__ mx,
                                   float* __restrict__ el, float* __restrict__ Z, int E) {
  int i = blockIdx.x * blockDim.x + threadIdx.x;
  if (i >= E) return;
  int t = dst[i];
  float e = __expf(logit[i] - mx[t]);
  el[i] = e;
  atomicAdd(Z + t, e);
}

// ---------------------------------------------------------------------------
// Edge pass 3 (heavy): fused radial MLP (WMMA) + tensor-product messages +
// attention-weighted scatter-add. One wave = 16 edges, 2 waves/block.
// Radial weights transposed in LDS for single-b64 B fragments.
// ---------------------------------------------------------------------------
#define EW 2
__global__ __launch_bounds__(EW * 32)
void edge_messages_kernel(const float* __restrict__ coords,
                          const int* __restrict__ src, const int* __restrict__ dst,
                          const float* __restrict__ efeat,
                          const float* __restrict__ el, const float* __restrict__ Z,
                          const float* __restrict__ h0n,
                          const float* __restrict__ h1x, const float* __restrict__ h1y,
                          const float* __restrict__ h1z,
                          const float* __restrict__ Wr1, const float* __restrict__ br1,
                          const float* __restrict__ Wr2,
                          float* __restrict__ a0,
                          float* __restrict__ a1x, float* __restrict__ a1y,
                          float* __restrict__ a1z,
                          int E, int has_f1) {
  __shared__ float sWr1t[32][38];   // [n][k], k<36 (rows 33..35 zero)
  __shared__ float sbr1[32];
  __shared__ float sWr2t[160][34];  // [n][k]
  __shared__ float sHid[EW][16][34];
  __shared__ float sRad[EW][16][164];
  __shared__ float sUx[EW][16], sUy[EW][16], sUz[EW][16], sAl[EW][16], sD[EW][16];
  __shared__ int   sS[EW][16], sT[EW][16];

  int tid = threadIdx.x;
  for (int idx = tid; idx < 32 * 36; idx += blockDim.x) {
    int n = idx / 36, k = idx % 36;
    sWr1t[n][k] = (k < 33) ? Wr1[k * 32 + n] : 0.f;
  }
  if (tid < 32) sbr1[tid] = br1[tid];
  for (int idx = tid; idx < 160 * 32; idx += blockDim.x) {
    int n = idx >> 5, k = idx & 31;
    sWr2t[n][k] = Wr2[k * 160 + n];
  }

  int wave = tid >> 5, lane = tid & 31;
  int e0 = (blockIdx.x * EW + wave) << 4;

  if (lane < 16) {
    int gi = e0 + lane;
    int s = src[gi], t = dst[gi];
    float rx = coords[t * 3 + 0] - coords[s * 3 + 0];
    float ry = coords[t * 3 + 1] - coords[s * 3 + 1];
    float rz = coords[t * 3 + 2] - coords[s * 3 + 2];
    float d  = sqrtf(rx * rx + ry * ry + rz * rz) + 1e-6f;
    float iv = 1.0f / d;
    sUx[wave][lane] = rx * iv; sUy[wave][lane] = ry * iv; sUz[wave][lane] = rz * iv;
    sD[wave][lane]  = d;
    sAl[wave][lane] = el[gi] / (Z[t] + 1e-9f);
    sS[wave][lane]  = s; sT[wave][lane] = t;
  }
  __syncthreads();

  int m = lane & 15, kh = lane >> 4;

  // Radial layer 1: [16 x 36(pad)] @ [36 x 32] -> relu -> sHid
  v8f h0acc = zero8();
  v8f h1acc = zero8();
#pragma unroll
  for (int kb = 0; kb < 36; kb += 4) {
    int k0 = kb + 2 * kh;
    v2f a;
    if (kb < 32) {
      a = *(const v2f*)(efeat + (size_t)(e0 + m) * 32 + k0);
    } else {
      a.x = (k0 == 32) ? sD[wave][m] : 0.f;
      a.y = 0.f;
    }
    v2f b0 = *(const v2f*)&sWr1t[m][k0];
    v2f b1 = *(const v2f*)&sWr1t[16 + m][k0];
    h0acc = wmma_f32(a, b0, h0acc);
    h1acc = wmma_f32(a, b1, h1acc);
  }
#pragma unroll
  for (int r = 0; r < 8; ++r) {
    int row = r + (kh << 3);
    sHid[wave][row][m]      = fmaxf(h0acc[r] + sbr1[m], 0.f);
    sHid[wave][row][16 + m] = fmaxf(h1acc[r] + sbr1[16 + m], 0.f);
  }
  __syncthreads();

  // Radial layer 2: [16 x 32] @ [32 x 160] -> sRad
  v8f rd[10];
#pragma unroll
  for (int t = 0; t < 10; ++t) rd[t] = zero8();
#pragma unroll
  for (int kb = 0; kb < 32; kb += 4) {
    int k0 = kb + 2 * kh;
    v2f a = *(const v2f*)&sHid[wave][m][k0];
#pragma unroll
    for (int t = 0; t < 10; ++t) {
      v2f b = *(const v2f*)&sWr2t[(t << 4) + m][k0];
      rd[t] = wmma_f32(a, b, rd[t]);
    }
  }
#pragma unroll
  for (int t = 0; t < 10; ++t) {
    int n = (t << 4) + m;
#pragma unroll
    for (int r = 0; r < 8; ++r) sRad[wave][r + (kh << 3)][n] = rd[t][r];
  }
  __syncthreads();

  // Tensor-product messages; lane = channel.
  int c = lane;
  for (int e = 0; e < 16; ++e) {
    int s = sS[wave][e], t = sT[wave][e];
    float al = sAl[wave][e];
    float ux = sUx[wave][e], uy = sUy[wave][e], uz = sUz[wave][e];
    size_t sb = (size_t)s * CCH + c;
    float h0v = h0n[sb];
    float hx = 0.f, hy = 0.f, hz = 0.f;
    if (has_f1) { hx = h1x[sb]; hy = h1y[sb]; hz = h1z[sb]; }
    float dt = hx * ux + hy * uy + hz * uz;
    float cx = hy * uz - hz * uy;
    float cy = hz * ux - hx * uz;
    float cz = hx * uy - hy * ux;
    float r0v = sRad[wave][e][c];
    float r1v = sRad[wave][e][32 + c];
    float r2v = sRad[wave][e][64 + c];
    float r3v = sRad[wave][e][96 + c];
    float r4v = sRad[wave][e][128 + c];
    float m0  = r0v * h0v + r1v * dt;
    float g   = r2v * h0v;
    float m1x = g * ux + r3v * hx + r4v * cx;
    float m1y = g * uy + r3v * hy + r4v * cy;
    float m1z = g * uz + r3v * hz + r4v * cz;
    size_t tb = (size_t)t * CCH + c;
    atomicAdd(a0  + tb, al * m0);
    atomicAdd(a1x + tb, al * m1x);
    atomicAdd(a1y + tb, al * m1y);
    atomicAdd(a1z + tb, al * m1z);
  }
}

// ---------------------------------------------------------------------------
// Output heads: tanh / relu projections + interleave l=1 vectors.
// ---------------------------------------------------------------------------
__global__ void heads_kernel(const float* __restrict__ f0,
                             const float* __restrict__ vx, const float* __restrict__ vy,
                             const float* __restrict__ vz,
                             const float* __restrict__ Wwo, const float* __restrict__ bwo,
                             const float* __restrict__ Wwb,
                             const float* __restrict__ Wsig, const float* __restrict__ bsig,
                             const float* __restrict__ Wc,
                             float* __restrict__ out, int n) {
  __shared__ float sWwo[32][8], sWwb[32][8], sWsig[32], sWc[32][10], sbwo[8], sbsig[1];
  int tid = threadIdx.x;
  for (int idx = tid; idx < 32 * 8; idx += blockDim.x) {
    sWwo[idx >> 3][idx & 7] = Wwo[idx];
    sWwb[idx >> 3][idx & 7] = Wwb[idx];
  }
  for (int idx = tid; idx < 32 * 10; idx += blockDim.x) sWc[idx / 10][idx % 10] = Wc[idx];
  if (tid < 32) sWsig[tid] = Wsig[tid];
  if (tid < 8)  sbwo[tid]  = bwo[tid];
  if (tid == 0) sbsig[0]   = bsig[0];
  __syncthreads();

  int i = blockIdx.x * blockDim.x + tid;
  if (i >= n) return;
  float f[32];
  const float4* fp = (const float4*)(f0 + (size_t)i * 32);
#pragma unroll
  for (int j = 0; j < 8; ++j) {
    float4 v = fp[j];
    f[4 * j] = v.x; f[4 * j + 1] = v.y; f[4 * j + 2] = v.z; f[4 * j + 3] = v.w;
  }
  float* wo = out;
  float* wb = out + (size_t)n * 8;
  float* cc = out + (size_t)n * 16;
  float* vv = out + (size_t)n * 26;
  float* sg = out + (size_t)n * 50;

#pragma unroll
  for (int o = 0; o < 8; ++o) {
    float s1 = 0.f, s2 = 0.f;
#pragma unroll
    for (int k = 0; k < 32; ++k) { s1 += f[k] * sWwo[k][o]; s2 += f[k] * sWwb[k][o]; }
    wo[(size_t)i * 8 + o] = tanhf(s1 + sbwo[o]);
    wb[(size_t)i * 8 + o] = tanhf(s2);
  }
#pragma unroll
  for (int o = 0; o < 10; ++o) {
    float s = 0.f;
#pragma unroll
    for (int k = 0; k < 32; ++k) s += f[k] * sWc[k][o];
    cc[(size_t)i * 10 + o] = fmaxf(s, 0.f);
  }
  {
    float s = 0.f;
#pragma unroll
    for (int k = 0; k < 32; ++k) s += f[k] * sWsig[k];
    sg[i] = fmaxf(s + sbsig[0], 0.f);
  }
#pragma unroll
  for (int o = 0; o < 8; ++o) {
    size_t b = (size_t)i * 24 + (size_t)o * 3;
    vv[b + 0] = vx[(size_t)i * 8 + o];
    vv[b + 1] = vy[(size_t)i * 8 + o];
    vv[b + 2] = vz[(size_t)i * 8 + o];
  }
}

// ---------------------------------------------------------------------------
extern "C" void kernel_launch(void* const* d_in, const int* in_sizes, int n_in,
                              void* d_out, int out_size, void* d_ws, size_t ws_size,
                              hipStream_t stream) {
  const float* node_f0 = (const float*)d_in[0];
  const float* coords  = (const float*)d_in[1];
  const float* efeat   = (const float*)d_in[2];
  const int*   esrc    = (const int*)d_in[3];
  const int*   edst    = (const int*)d_in[4];
  // params1: Wr1, br1, Wr2, W0, W1, Wq, Wk, Wo0, Wo1, Ws0
  const float* p1[10]; for (int i = 0; i < 10; ++i) p1[i] = (const float*)d_in[5 + i];
  const float* p2[10]; for (int i = 0; i < 10; ++i) p2[i] = (const float*)d_in[15 + i];
  const float* hWwo  = (const float*)d_in[25];
  const float* hbwo  = (const float*)d_in[26];
  const float* hWwb  = (const float*)d_in[27];
  const float* hWsig = (const float*)d_in[28];
  const float* hbsig = (const float*)d_in[29];
  const float* hWc   = (const float*)d_in[30];

  // Workspace layout
  float* w = (float*)d_ws;
  size_t off = 0;
  auto take = [&](size_t n) { float* p = w + off; off += n; return p; };
  float* h0n  = take((size_t)NN * 32);
  float* qn   = take((size_t)NN * 32);
  float* kn   = take((size_t)NN * 32);
  float* h1x  = take((size_t)NN * 32);
  float* h1y  = take((size_t)NN * 32);
  float* h1z  = take((size_t)NN * 32);
  float* f1x  = take((size_t)NN * 32);
  float* f1y  = take((size_t)NN * 32);
  float* f1z  = take((size_t)NN * 32);
  float* a0   = take((size_t)NN * 32);
  float* a1x  = take((size_t)NN * 32);
  float* a1y  = take((size_t)NN * 32);
  float* a1z  = take((size_t)NN * 32);
  float* f0a  = take((size_t)NN * 32);
  float* f0b  = take((size_t)NN * 32);
  float* vx   = take((size_t)NN * 8);
  float* vy   = take((size_t)NN * 8);
  float* vz   = take((size_t)NN * 8);
  float* mx   = take(NN);
  float* Zs   = take(NN);
  float* logit = take(NE);
  float* el    = take(NE);

  const int nodeBlocks = (NN + 255) / 256;
  const int edgeBlocks = (NE + 255) / 256;
  const int msgBlocks  = NE / (EW * 16);

  auto gemm = [&](const float* X, const float* W, float* Y, int M, int NO, int acc) {
    gemm_wmma_kernel<<<(M + 63) / 64, 128, 0, stream>>>(X, W, Y, M, NO, acc);
  };
  (void)in_sizes; (void)n_in; (void)out_size; (void)ws_size;

  // ================= Layer 1 (f1 input == 0) =================
  prep_accum_kernel<<<nodeBlocks, 256, 0, stream>>>(mx, Zs, a0, a1x, a1y, a1z, NN);
  gemm(node_f0, p1[3], h0n, NN, 32, 0);   // W0
  gemm(node_f0, p1[5], qn,  NN, 32, 0);   // Wq
  gemm(node_f0, p1[6], kn,  NN, 32, 0);   // Wk
  edge_logits_kernel<<<edgeBlocks, 256, 0, stream>>>(esrc, edst, qn, kn, logit, mx, NE);
  edge_expsum_kernel<<<edgeBlocks, 256, 0, stream>>>(edst, logit, mx, el, Zs, NE);
  edge_messages_kernel<<<msgBlocks, EW * 32, 0, stream>>>(
      coords, esrc, edst, efeat, el, Zs, h0n,
      nullptr, nullptr, nullptr,
      p1[0], p1[1], p1[2],
      a0, a1x, a1y, a1z, NE, 0);
  gemm(a0,      p1[7], f0a, NN, 32, 0);   // Wo0
  gemm(node_f0, p1[9], f0a, NN, 32, 1);   // + Ws0
  gemm(a1x, p1[8], f1x, NN, 32, 0);       // Wo1
  gemm(a1y, p1[8], f1y, NN, 32, 0);
  gemm(a1z, p1[8], f1z, NN, 32, 0);

  // ================= Layer 2 =================
  prep_accum_kernel<<<nodeBlocks, 256, 0, stream>>>(mx, Zs, a0, a1x, a1y, a1z, NN);
  gemm(f0a, p2[3], h0n, NN, 32, 0);       // W0
  gemm(f0a, p2[5], qn,  NN, 32, 0);       // Wq
  gemm(f0a, p2[6], kn,  NN, 32, 0);       // Wk
  gemm(f1x, p2[4], h1x, NN, 32, 0);       // W1
  gemm(f1y, p2[4], h1y, NN, 32, 0);
  gemm(f1z, p2[4], h1z, NN, 32, 0);
  edge_logits_kernel<<<edgeBlocks, 256, 0, stream>>>(esrc, edst, qn, kn, logit, mx, NE);
  edge_expsum_kernel<<<edgeBlocks, 256, 0, stream>>>(edst, logit, mx, el, Zs, NE);
  edge_messages_kernel<<<msgBlocks, EW * 32, 0, stream>>>(
      coords, esrc, edst, efeat, el, Zs, h0n,
      h1x, h1y, h1z,
      p2[0], p2[1], p2[2],
      a0, a1x, a1y, a1z, NE, 1);
  gemm(a0,  p2[7], f0b, NN, 32, 0);       // Wo0
  gemm(f0a, p2[9], f0b, NN, 32, 1);       // + Ws0
  gemm(a1x, p2[8], vx, NN, 8, 0);         // Wo1 (cout1=8)
  gemm(a1y, p2[8], vy, NN, 8, 0);
  gemm(a1z, p2[8], vz, NN, 8, 0);

  // ================= Heads =================
  heads_kernel<<<nodeBlocks, 256, 0, stream>>>(
      f0b, vx, vy, vz, hWwo, hbwo, hWwb, hWsig, hbsig, hWc, (float*)d_out, NN);
}